// CircuitGNN_Polarity_32384053412274
// MI455X (gfx1250) — compile-verified
//
#include <hip/hip_runtime.h>
#include <hip/hip_bf16.h>

typedef __attribute__((ext_vector_type(2))) float v2f;
typedef __attribute__((ext_vector_type(8))) float v8f;

#define NEG_SLOPE 0.2f
#define BN_EPS 1e-5f

// float atomic max via sign-aware integer atomics (correct for non-NaN,
// initial value -1e30f). Lowers to native global integer atomics.
__device__ __forceinline__ void atomicMaxF(float* addr, float val) {
    if (val >= 0.0f) atomicMax((int*)addr, __float_as_int(val));
    else             atomicMin((unsigned int*)addr, __float_as_uint(val));
}

// -------------------------------------------------------------------------
// Node transform: [XL | XR] = X @ [Wl | Wr], via V_WMMA_F32_16X16X4_F32.
// One block = one 16-node row tile. Each wave32 owns one 16-column tile of
// the concatenated 2*NC output columns. Both operands are staged in LDS,
// zero-padded to KPAD, so the unrolled WMMA chain has no guards and no
// divergence. B rows are stride-padded (+16 floats) so the two half-waves
// (rows kk / kk+2) hit disjoint LDS bank groups.
// A frag (16x4 f32): lane<16 -> {A[M][k0],A[M][k0+1]}, lane>=16 -> k0+2,k0+3.
// D layout: VGPR r: lanes 0-15 = D[r][n], lanes 16-31 = D[r+8][n].
// Store epilogue: uniform branch selects an unguarded fast path for full
// row tiles (always taken when n % 16 == 0).
// -------------------------------------------------------------------------
template <int KPAD, int NC>
__global__ void gat_transform(const float* __restrict__ X, int ldx, int K,
                              const float* __restrict__ Wl,
                              const float* __restrict__ Wr,
                              float* __restrict__ XL, float* __restrict__ XR,
                              int n) {
    constexpr int NITER   = KPAD / 4;
    constexpr int NC2     = 2 * NC;
    constexpr int BSTRIDE = NC2 + 16;
    __shared__ float tA[16][KPAD + 1];
    __shared__ float tB[KPAD][BSTRIDE];

    const int base = blockIdx.x * 16;

    // Stage A tile [16 x KPAD], zero-padded.
    for (int i = threadIdx.x; i < 16 * KPAD; i += blockDim.x) {
        const int r = i / KPAD, k = i - r * KPAD;
        const int row = base + r;
        tA[r][k] = (row < n && k < K) ? X[(size_t)row * ldx + k] : 0.0f;
    }
    // Stage B = [Wl | Wr] tile [KPAD x 2NC], zero-padded rows K..KPAD-1.
    for (int i = threadIdx.x; i < KPAD * NC2; i += blockDim.x) {
        const int k = i / NC2, c = i - k * NC2;
        float w = 0.0f;
        if (k < K) w = (c < NC) ? Wl[(size_t)k * NC + c] : Wr[(size_t)k * NC + (c - NC)];
        tB[k][c] = w;
    }
    __syncthreads();

    const int wave = threadIdx.x >> 5;
    const int lane = threadIdx.x & 31;
    const int l15  = lane & 15;
    const int kh   = (lane >> 4) << 1;     // 0 or 2
    const int cg   = wave * 16 + l15;      // column in concatenated [Wl|Wr]

    // Preload all fragments into registers, then run the WMMA chain
    // back-to-back on the matrix pipe.
    v2f a[NITER], b[NITER];
#pragma unroll
    for (int i = 0; i < NITER; ++i) {
        const int kk = i * 4 + kh;
        a[i].x = tA[l15][kk];
        a[i].y = tA[l15][kk + 1];
        b[i].x = tB[kk][cg];
        b[i].y = tB[kk + 1][cg];
    }
    v8f acc = {};
#pragma unroll
    for (int i = 0; i < NITER; ++i)
        acc = __builtin_amdgcn_wmma_f32_16x16x4_f32(
            /*neg_a=*/false, a[i], /*neg_b=*/false, b[i],
            /*c_mod=*/(short)0, acc, /*reuse_a=*/false, /*reuse_b=*/false);

    float* __restrict__ Y = (wave < NC / 16) ? XL : XR;
    const int c0   = (wave < NC / 16) ? (wave << 4) : ((wave - NC / 16) << 4);
    const int Mofs = (lane < 16) ? 0 : 8;
    if (base + 16 <= n) {
        // Full tile: uniform, unguarded stores.
#pragma unroll
        for (int r = 0; r < 8; ++r)
            Y[(size_t)(base + Mofs + r) * NC + c0 + l15] = acc[r];
    } else {
        // Tail tile: per-row guard.
#pragma unroll
        for (int r = 0; r < 8; ++r) {
            const int row = base + Mofs + r;
            if (row < n) Y[(size_t)row * NC + c0 + l15] = acc[r];
        }
    }
}

// Per-layer reset of softmax state + aggregation buffer.
__global__ void gat_init(float* __restrict__ smax, float* __restrict__ denom,
                         float* __restrict__ accum, int n, int NC) {
    const int stride = gridDim.x * blockDim.x;
    const int tid = blockIdx.x * blockDim.x + threadIdx.x;
    for (int k = tid; k < n * 2; k += stride) { smax[k] = -1e30f; denom[k] = 0.0f; }
    for (int k = tid; k < n * NC; k += stride) accum[k] = 0.0f;
}

// Pass A: per-edge attention score (both heads) + atomic per-dst max.
// One wave32 per edge, lane = channel (edge scalars are wave-uniform loads).
__global__ void edge_score(const float* __restrict__ xl, const float* __restrict__ xr,
                           const int* __restrict__ esrc, const int* __restrict__ edst,
                           const float* __restrict__ att, float* __restrict__ sbuf,
                           float* __restrict__ smax, int E, int Etot, int C) {
    const int eid  = (blockIdx.x * blockDim.x + threadIdx.x) >> 5;
    const int lane = threadIdx.x & 31;
    if (eid >= Etot) return;
    int src, dst;
    if (eid < E) { src = esrc[eid]; dst = edst[eid]; } else { src = dst = eid - E; }
    const int NC = 2 * C;
    float p0 = 0.0f, p1 = 0.0f;
    if (lane < C) {
        float v0 = xl[(size_t)src * NC + lane] + xr[(size_t)dst * NC + lane];
        v0 = (v0 > 0.0f) ? v0 : NEG_SLOPE * v0;
        p0 = v0 * att[lane];
        float v1 = xl[(size_t)src * NC + C + lane] + xr[(size_t)dst * NC + C + lane];
        v1 = (v1 > 0.0f) ? v1 : NEG_SLOPE * v1;
        p1 = v1 * att[C + lane];
    }
#pragma unroll
    for (int off = 16; off > 0; off >>= 1) {
        p0 += __shfl_xor(p0, off, 32);
        p1 += __shfl_xor(p1, off, 32);
    }
    if (lane == 0) {
        sbuf[(size_t)eid * 2]     = p0;
        sbuf[(size_t)eid * 2 + 1] = p1;
        atomicMaxF(&smax[(size_t)dst * 2], p0);
        atomicMaxF(&smax[(size_t)dst * 2 + 1], p1);
    }
}

// Pass B: p = exp(s - smax[dst]); atomic per-dst sum. One thread per (edge, head).
__global__ void edge_norm(float* __restrict__ sbuf, const float* __restrict__ smax,
                          float* __restrict__ denom,
                          const int* __restrict__ edst, int E, int Etot) {
    const int idx = blockIdx.x * blockDim.x + threadIdx.x;
    if (idx >= Etot * 2) return;
    const int eid = idx >> 1;
    const int h   = idx & 1;
    const int dst = (eid < E) ? edst[eid] : (eid - E);
    const float p = __expf(sbuf[idx] - smax[(size_t)dst * 2 + h]);
    sbuf[idx] = p;
    atomicAdd(&denom[(size_t)dst * 2 + h], p);
}

// Pass C: accum[dst] += (p/denom[dst]) * xl[src]. One wave32 per edge.
__global__ void edge_agg(const float* __restrict__ xl, const float* __restrict__ sbuf,
                         const float* __restrict__ denom,
                         const int* __restrict__ esrc, const int* __restrict__ edst,
                         float* __restrict__ accum, int E, int Etot, int C) {
    const int eid  = (blockIdx.x * blockDim.x + threadIdx.x) >> 5;
    const int lane = threadIdx.x & 31;
    if (eid >= Etot) return;
    int src, dst;
    if (eid < E) { src = esrc[eid]; dst = edst[eid]; } else { src = dst = eid - E; }
    const int NC = 2 * C;
    const float w0 = sbuf[(size_t)eid * 2]     / denom[(size_t)dst * 2];
    const float w1 = sbuf[(size_t)eid * 2 + 1] / denom[(size_t)dst * 2 + 1];
    if (lane < C) {
        atomicAdd(&accum[(size_t)dst * NC + lane],     w0 * xl[(size_t)src * NC + lane]);
        atomicAdd(&accum[(size_t)dst * NC + C + lane], w1 * xl[(size_t)src * NC + C + lane]);
    }
}

// head-mean + bias + BatchNorm + ELU (+ optional residual)
__global__ void gat_finalize(const float* __restrict__ accum, const float* __restrict__ bias,
                             const float* __restrict__ g, const float* __restrict__ bb,
                             const float* __restrict__ m, const float* __restrict__ v,
                             const float* __restrict__ resid, float* __restrict__ out,
                             int n, int C) {
    const int idx = blockIdx.x * blockDim.x + threadIdx.x;
    if (idx >= n * C) return;
    const int node = idx / C;
    const int c    = idx - node * C;
    const int NC   = 2 * C;
    float mean = 0.5f * (accum[(size_t)node * NC + c] + accum[(size_t)node * NC + C + c]) + bias[c];
    float y = (mean - m[c]) * rsqrtf(v[c] + BN_EPS) * g[c] + bb[c];
    y = (y > 0.0f) ? y : (__expf(y) - 1.0f);
    if (resid) y += resid[idx];
    out[idx] = y;
}

// sigmoid(h16 @ head_w + head_b)
__global__ void gat_head(const float* __restrict__ h16, const float* __restrict__ hw,
                         const float* __restrict__ hb, float* __restrict__ out, int n) {
    const int node = blockIdx.x * blockDim.x + threadIdx.x;
    if (node >= n) return;
    float z = hb[0];
#pragma unroll
    for (int c = 0; c < 16; ++c) z += h16[(size_t)node * 16 + c] * hw[c];
    out[node] = 1.0f / (1.0f + __expf(-z));
}

extern "C" void kernel_launch(void* const* d_in, const int* in_sizes, int n_in,
                              void* d_out, int out_size, void* d_ws, size_t ws_size,
                              hipStream_t stream) {
    const float* x    = (const float*)d_in[0];
    const int*   esrc = (const int*)d_in[1];
    const int*   edst = (const int*)d_in[2];
    const float* W0_l = (const float*)d_in[3];
    const float* W0_r = (const float*)d_in[4];
    const float* att0 = (const float*)d_in[5];
    const float* b0   = (const float*)d_in[6];
    const float* bn0g = (const float*)d_in[7];
    const float* bn0b = (const float*)d_in[8];
    const float* bn0m = (const float*)d_in[9];
    const float* bn0v = (const float*)d_in[10];
    const float* Wm_l = (const float*)d_in[11];
    const float* Wm_r = (const float*)d_in[12];
    const float* attm = (const float*)d_in[13];
    const float* bm   = (const float*)d_in[14];
    const float* bnmg = (const float*)d_in[15];
    const float* bnmb = (const float*)d_in[16];
    const float* bnmm = (const float*)d_in[17];
    const float* bnmv = (const float*)d_in[18];
    const float* W9_l = (const float*)d_in[19];
    const float* W9_r = (const float*)d_in[20];
    const float* att9 = (const float*)d_in[21];
    const float* b9   = (const float*)d_in[22];
    const float* bn9g = (const float*)d_in[23];
    const float* bn9b = (const float*)d_in[24];
    const float* bn9m = (const float*)d_in[25];
    const float* bn9v = (const float*)d_in[26];
    const float* hw   = (const float*)d_in[27];
    const float* hb   = (const float*)d_in[28];

    const int N    = in_sizes[0] / 17;
    const int E    = in_sizes[1];
    const int Etot = E + N;

    // workspace layout (floats)
    float* ws    = (float*)d_ws;
    float* h_cur = ws;                                 // N*32
    float* h_nxt = h_cur + (size_t)N * 32;             // N*32
    float* xl    = h_nxt + (size_t)N * 32;             // N*64
    float* xr    = xl    + (size_t)N * 64;             // N*64
    float* accum = xr    + (size_t)N * 64;             // N*64
    float* smax  = accum + (size_t)N * 64;             // N*2
    float* denom = smax  + (size_t)N * 2;              // N*2
    float* sbuf  = denom + (size_t)N * 2;              // Etot*2
    float* h16   = sbuf  + (size_t)Etot * 2;           // N*16

    const int rowTiles   = (N + 15) / 16;
    const int edgeBlocks = (Etot * 32 + 255) / 256;    // wave-per-edge kernels

    auto run_edges = [&](const float* att, const float* bias, const float* g,
                         const float* bb, const float* m, const float* v,
                         const float* resid, float* hout, int C) {
        const int NC = 2 * C;
        gat_init<<<(N * NC + 255) / 256, 256, 0, stream>>>(smax, denom, accum, N, NC);
        edge_score<<<edgeBlocks, 256, 0, stream>>>(xl, xr, esrc, edst, att, sbuf, smax, E, Etot, C);
        edge_norm<<<(Etot * 2 + 255) / 256, 256, 0, stream>>>(sbuf, smax, denom, edst, E, Etot);
        edge_agg<<<edgeBlocks, 256, 0, stream>>>(xl, sbuf, denom, esrc, edst, accum, E, Etot, C);
        gat_finalize<<<(N * C + 255) / 256, 256, 0, stream>>>(accum, bias, g, bb, m, v, resid, hout, N, C);
    };

    // ---- layer 0: 17 -> 32, K padded 17 -> 20 ----
    gat_transform<20, 32><<<rowTiles, 256, 0, stream>>>(x, 17, 17, W0_l, W0_r, xl, xr, N);
    run_edges(att0, b0, bn0g, bn0b, bn0m, bn0v, nullptr, h_cur, 32);

    // ---- layers 1..8: 32 -> 32 with residual ----
    for (int l = 0; l < 8; ++l) {
        const float* Wl = Wm_l + (size_t)l * 32 * 64;
        const float* Wr = Wm_r + (size_t)l * 32 * 64;
        gat_transform<32, 32><<<rowTiles, 256, 0, stream>>>(h_cur, 32, 32, Wl, Wr, xl, xr, N);
        run_edges(attm + (size_t)l * 64, bm + (size_t)l * 32,
                  bnmg + (size_t)l * 32, bnmb + (size_t)l * 32,
                  bnmm + (size_t)l * 32, bnmv + (size_t)l * 32,
                  h_cur, h_nxt, 32);
        float* t = h_cur; h_cur = h_nxt; h_nxt = t;
    }

    // ---- layer 9: 32 -> 16 (per-output NC = 32 -> 4 waves/block) ----
    gat_transform<32, 16><<<rowTiles, 128, 0, stream>>>(h_cur, 32, 32, W9_l, W9_r, xl, xr, N);
    run_edges(att9, b9, bn9g, bn9b, bn9m, bn9v, nullptr, h16, 16);

    // ---- sigmoid head ----
    gat_head<<<(N + 255) / 256, 256, 0, stream>>>(h16, hw, hb, (float*)d_out, N);
}